// GCN_82360292868212
// MI455X (gfx1250) — compile-verified
//
#include <hip/hip_runtime.h>
#include <cstdint>

typedef __attribute__((ext_vector_type(2))) float v2f;
typedef __attribute__((ext_vector_type(4))) float v4f;
typedef __attribute__((ext_vector_type(8))) float v8f;

// ---------------------------------------------------------------------------
// Zero the first nWords4 * 4 dwords of workspace (deg_src, deg_dst, agg).
// ---------------------------------------------------------------------------
__global__ void k_zero(float* __restrict__ p, int nWords4) {
    int i = blockIdx.x * blockDim.x + threadIdx.x;
    if (i < nWords4) {
        v4f z = {0.f, 0.f, 0.f, 0.f};
        ((v4f*)p)[i] = z;
    }
}

// ---------------------------------------------------------------------------
// Degree counting: one thread per edge, u32 atomics into L2-resident tables.
// ---------------------------------------------------------------------------
__global__ void k_deg(const int* __restrict__ src, const int* __restrict__ dst,
                      unsigned* __restrict__ deg_src, unsigned* __restrict__ deg_dst, int nE) {
    int e = blockIdx.x * blockDim.x + threadIdx.x;
    if (e < nE) {
        (void)__hip_atomic_fetch_add(&deg_src[src[e]], 1u, __ATOMIC_RELAXED, __HIP_MEMORY_SCOPE_AGENT);
        (void)__hip_atomic_fetch_add(&deg_dst[dst[e]], 1u, __ATOMIC_RELAXED, __HIP_MEMORY_SCOPE_AGENT);
    }
}

// ---------------------------------------------------------------------------
// In-place deg -> deg^{-1/2} (0 for isolated nodes), for both tables.
// ---------------------------------------------------------------------------
__global__ void k_norm(float* __restrict__ ns, float* __restrict__ nd, int nN) {
    int i = blockIdx.x * blockDim.x + threadIdx.x;
    if (i < nN) {
        unsigned ds = ((const unsigned*)ns)[i];
        unsigned dd = ((const unsigned*)nd)[i];
        ns[i] = ds ? rsqrtf((float)ds) : 0.f;
        nd[i] = dd ? rsqrtf((float)dd) : 0.f;
    }
}

// ---------------------------------------------------------------------------
// Layer-1 projection: in_feat is [N,1] so h = (x*norm_src) * W1_row (1x4).
// ---------------------------------------------------------------------------
__global__ void k_h1(const float* __restrict__ x, const float* __restrict__ ns,
                     const float* __restrict__ W1, float* __restrict__ h, int nN) {
    int i = blockIdx.x * blockDim.x + threadIdx.x;
    if (i < nN) {
        float f = x[i] * ns[i];
        v4f o = {f * W1[0], f * W1[1], f * W1[2], f * W1[3]};
        ((v4f*)h)[i] = o;
    }
}

// ---------------------------------------------------------------------------
// Edge scatter: agg[dst] += h[src]. Gather is a b128 load (L2 resident),
// scatter is 4 relaxed no-return f32 atomics (global_atomic_add_f32).
// ---------------------------------------------------------------------------
__global__ void k_scatter(const int* __restrict__ src, const int* __restrict__ dst,
                          const float* __restrict__ h, float* agg, int nE) {
    int e = blockIdx.x * blockDim.x + threadIdx.x;
    if (e < nE) {
        int s = src[e];
        int d = dst[e];
        v4f v = ((const v4f*)h)[s];
        float* ap = agg + (size_t)d * 4;
        (void)__hip_atomic_fetch_add(ap + 0, v.x, __ATOMIC_RELAXED, __HIP_MEMORY_SCOPE_AGENT);
        (void)__hip_atomic_fetch_add(ap + 1, v.y, __ATOMIC_RELAXED, __HIP_MEMORY_SCOPE_AGENT);
        (void)__hip_atomic_fetch_add(ap + 2, v.z, __ATOMIC_RELAXED, __HIP_MEMORY_SCOPE_AGENT);
        (void)__hip_atomic_fetch_add(ap + 3, v.w, __ATOMIC_RELAXED, __HIP_MEMORY_SCOPE_AGENT);
    }
}

// ---------------------------------------------------------------------------
// Fused finalize(prev layer) + project(next layer) via V_WMMA_F32_16X16X4_F32.
//   x   = relu(agg * norm_dst + bias)        (prev layer epilogue)
//   a   = x * norm_src                        (next layer pre-scale)
//   h   = a @ W                               (16 nodes per WMMA, K=4)
// Also re-zeroes agg for the next scatter round.
// A-matrix layout (16x4 f32): lanes 0-15 hold K=0,1; lanes 16-31 hold K=2,3
// -> each lane loads a v2f of its node's features (coalesced 256B/wave).
// B-matrix (4x16): VGPR0 = rows K=0 (lanes 0-15) / K=2 (lanes 16-31),
//                  VGPR1 = rows K=1 / K=3; columns >= 4 zero-padded.
// D (16x16 f32): VGPR r = row r (lanes 0-15, N=lane) / row r+8 (lanes 16-31).
// 128 threads = 4 waves = 64 nodes per block; EXEC all-ones through the WMMA.
// ---------------------------------------------------------------------------
__global__ void __launch_bounds__(128) k_proj_wmma(
        float* agg, const float* __restrict__ ns, const float* __restrict__ nd,
        const float* __restrict__ W, const float* __restrict__ bias,
        float* __restrict__ h) {
    const int lane = threadIdx.x & 31;
    const int wv   = threadIdx.x >> 5;
    const int tile = blockIdx.x * 4 + wv;     // 16 nodes per tile
    const int m    = lane & 15;               // node-in-tile / column index
    const int hi   = lane >> 4;               // lane half selects K pair
    const int node = tile * 16 + m;
    const int kb   = hi * 2;                  // feature base: 0 or 2

    // A fragment: finalize previous layer then pre-scale
    v2f acc = *(const v2f*)(agg + (size_t)node * 4 + kb);
    float sd = nd[node];
    float ss = ns[node];
    float x0 = fmaxf(acc.x * sd + bias[kb],     0.f);
    float x1 = fmaxf(acc.y * sd + bias[kb + 1], 0.f);
    v2f a;
    a.x = x0 * ss;
    a.y = x1 * ss;

    // re-zero agg for the next scatter round
    v2f z2 = {0.f, 0.f};
    *(v2f*)(agg + (size_t)node * 4 + kb) = z2;

    // B fragment: W is 4x4 row-major (W[k][n] = W[k*4+n]); pad cols >= 4
    v2f bf;
    bf.x = (m < 4) ? W[kb * 4 + m]       : 0.f;
    bf.y = (m < 4) ? W[(kb + 1) * 4 + m] : 0.f;

    v8f c = {};
    v8f dm = __builtin_amdgcn_wmma_f32_16x16x4_f32(
        /*neg_a=*/false, a, /*neg_b=*/false, bf,
        /*c_mod=*/(short)0, c, /*reuse_a=*/false, /*reuse_b=*/false);

    // Store: only columns 0..3 are real outputs. Lane n holds h[row][n] for
    // rows r (+8 for upper half) across the 8 D VGPRs.
    if (m < 4) {
        float* hp = h + ((size_t)tile * 16 + hi * 8) * 4 + m;
        hp[0]  = dm[0];
        hp[4]  = dm[1];
        hp[8]  = dm[2];
        hp[12] = dm[3];
        hp[16] = dm[4];
        hp[20] = dm[5];
        hp[24] = dm[6];
        hp[28] = dm[7];
    }
}

// VALU fallback for a non-multiple-of-64 node tail (not hit for N = 1M).
__global__ void k_proj_tail(float* agg, const float* __restrict__ ns,
                            const float* __restrict__ nd, const float* __restrict__ W,
                            const float* __restrict__ bias, float* __restrict__ h,
                            int start, int nN) {
    int i = start + blockIdx.x * blockDim.x + threadIdx.x;
    if (i < nN) {
        v4f acc = ((const v4f*)agg)[i];
        float sd = nd[i], ss = ns[i];
        float x0 = fmaxf(acc.x * sd + bias[0], 0.f) * ss;
        float x1 = fmaxf(acc.y * sd + bias[1], 0.f) * ss;
        float x2 = fmaxf(acc.z * sd + bias[2], 0.f) * ss;
        float x3 = fmaxf(acc.w * sd + bias[3], 0.f) * ss;
        v4f z = {0.f, 0.f, 0.f, 0.f};
        ((v4f*)agg)[i] = z;
        v4f o;
        o.x = x0 * W[0] + x1 * W[4] + x2 * W[8]  + x3 * W[12];
        o.y = x0 * W[1] + x1 * W[5] + x2 * W[9]  + x3 * W[13];
        o.z = x0 * W[2] + x1 * W[6] + x2 * W[10] + x3 * W[14];
        o.w = x0 * W[3] + x1 * W[7] + x2 * W[11] + x3 * W[15];
        ((v4f*)h)[i] = o;
    }
}

// ---------------------------------------------------------------------------
// Final: x3 = agg*norm_dst + b3 ; logits = x3 @ Wl + bl ; 2-class softmax.
// ---------------------------------------------------------------------------
__global__ void k_final(const float* __restrict__ agg, const float* __restrict__ nd,
                        const float* __restrict__ b3, const float* __restrict__ Wl,
                        const float* __restrict__ bl, float* __restrict__ out, int nN) {
    int i = blockIdx.x * blockDim.x + threadIdx.x;
    if (i < nN) {
        v4f acc = ((const v4f*)agg)[i];
        float s  = nd[i];
        float x0 = acc.x * s + b3[0];
        float x1 = acc.y * s + b3[1];
        float x2 = acc.z * s + b3[2];
        float x3 = acc.w * s + b3[3];
        // Wl is (4,2) row-major: Wl[j][k] = Wl[j*2+k]
        float l0 = bl[0] + x0 * Wl[0] + x1 * Wl[2] + x2 * Wl[4] + x3 * Wl[6];
        float l1 = bl[1] + x0 * Wl[1] + x1 * Wl[3] + x2 * Wl[5] + x3 * Wl[7];
        float mx = fmaxf(l0, l1);
        float e0 = __expf(l0 - mx);
        float e1 = __expf(l1 - mx);
        float inv = 1.f / (e0 + e1);
        out[(size_t)i * 2]     = e0 * inv;
        out[(size_t)i * 2 + 1] = e1 * inv;
    }
}

extern "C" void kernel_launch(void* const* d_in, const int* in_sizes, int n_in,
                              void* d_out, int out_size, void* d_ws, size_t ws_size,
                              hipStream_t stream) {
    (void)n_in; (void)out_size; (void)ws_size;
    const float* in_feat = (const float*)d_in[0];
    const int*   src     = (const int*)d_in[1];
    const int*   dst     = (const int*)d_in[2];
    const float* W1      = (const float*)d_in[3];
    const float* b1      = (const float*)d_in[4];
    const float* W2      = (const float*)d_in[5];
    const float* b2      = (const float*)d_in[6];
    const float* W3      = (const float*)d_in[7];
    const float* b3      = (const float*)d_in[8];
    const float* Wl      = (const float*)d_in[9];
    const float* bl      = (const float*)d_in[10];
    float*       out     = (float*)d_out;

    const int nN = in_sizes[0];   // 1,000,000
    const int nE = in_sizes[1];   // 8,000,000

    // Workspace layout (floats): [norm_src nN][norm_dst nN][agg 4*nN][h 4*nN]
    float* ws       = (float*)d_ws;
    float* norm_src = ws;
    float* norm_dst = ws + (size_t)nN;
    float* agg      = ws + (size_t)2 * nN;
    float* h        = ws + (size_t)6 * nN;

    const int B = 256;
    const int gN = (nN + B - 1) / B;
    const int gE = (nE + B - 1) / B;

    // 1) zero deg_src, deg_dst, agg (first 6*nN dwords), 16B per thread
    int zero4 = (6 * nN + 3) / 4;
    k_zero<<<(zero4 + B - 1) / B, B, 0, stream>>>(ws, zero4);

    // 2) degrees + normalization constants (shared by all 3 layers)
    k_deg<<<gE, B, 0, stream>>>(src, dst, (unsigned*)norm_src, (unsigned*)norm_dst, nE);
    k_norm<<<gN, B, 0, stream>>>(norm_src, norm_dst, nN);

    // 3) layer 1: project (K=1) -> scatter
    k_h1<<<gN, B, 0, stream>>>(in_feat, norm_src, W1, h, nN);
    k_scatter<<<gE, B, 0, stream>>>(src, dst, h, agg, nE);

    // 4) layer 2: fused relu-finalize(b1) + WMMA project(W2) -> scatter
    const int projBlocks = nN / 64;         // 64 nodes per 128-thread block
    const int tail       = projBlocks * 64;
    k_proj_wmma<<<projBlocks, 128, 0, stream>>>(agg, norm_src, norm_dst, W2, b1, h);
    if (tail < nN)
        k_proj_tail<<<(nN - tail + B - 1) / B, B, 0, stream>>>(agg, norm_src, norm_dst,
                                                               W2, b1, h, tail, nN);
    k_scatter<<<gE, B, 0, stream>>>(src, dst, h, agg, nE);

    // 5) layer 3: fused relu-finalize(b2) + WMMA project(W3) -> scatter
    k_proj_wmma<<<projBlocks, 128, 0, stream>>>(agg, norm_src, norm_dst, W3, b2, h);
    if (tail < nN)
        k_proj_tail<<<(nN - tail + B - 1) / B, B, 0, stream>>>(agg, norm_src, norm_dst,
                                                               W3, b2, h, tail, nN);
    k_scatter<<<gE, B, 0, stream>>>(src, dst, h, agg, nE);

    // 6) finalize(b3) + linear head + softmax
    k_final<<<gN, B, 0, stream>>>(agg, norm_dst, b3, Wl, bl, out, nN);
}